// LinearFlowModel_25786983646071
// MI455X (gfx1250) — compile-verified
//
#include <hip/hip_runtime.h>

// D = A(16x4, f32) x B(4x16, f32) + C(16x16, f32) via V_WMMA_F32_16X16X4_F32.
// out[b,(n,o)] = state[b,:] @ W[(n,o),:]^T + bias[(n,o)]
//   M = B rows (131072), K = 128, N = 256 logical cols, col = (n,o).
// W staged into LDS with rows reordered as r = o*128 + n, so N-tile group 0
// (tiles 0..7) -> delta and group 1 (tiles 8..15) -> var, coalesced stores.
// N-tiles processed in 2 groups of 8 to keep VGPR pressure < 256
// (no s_set_vgpr_msb, higher occupancy); A fragments stay resident so the
// 64 MiB state array is still read exactly once.

typedef __attribute__((ext_vector_type(2))) float v2f;
typedef __attribute__((ext_vector_type(4))) float v4f;
typedef __attribute__((ext_vector_type(8))) float v8f;

#define LDS_STRIDE 132   // 128 + 4 floats pad: ds_load_b64 B-frag pattern is bank-conflict-free

__global__ __launch_bounds__(256) void flow_gemm_f32_wmma(
    const float* __restrict__ state,   // (B,128)
    const float* __restrict__ W,       // (128,2,128) = W[n][o][d]
    const float* __restrict__ bias,    // (128,2)     = b[n][o]
    float* __restrict__ outDelta,      // (B,128)
    float* __restrict__ outVar,        // (B,128)
    int tiles)                         // B/16
{
    extern __shared__ float smem[];
    float* sW    = smem;                        // 256 rows x LDS_STRIDE floats
    float* sBias = smem + 256 * LDS_STRIDE;     // 256 floats, index = o*128+n

    // ---- Stage W into LDS, reordering rows: LDS row (o*128+n) = W[n][o][0..127]
    {
        const v4f* Wv = (const v4f*)W;          // 8192 float4s
        for (int i = threadIdx.x; i < 8192; i += 256) {
            v4f w4 = Wv[i];
            int f = i << 2;                     // flat float index = n*256 + o*128 + k
            int n = f >> 8;
            int o = (f >> 7) & 1;
            int k = f & 127;
            *(v4f*)&sW[(o * 128 + n) * LDS_STRIDE + k] = w4;   // 16B aligned (528B rows)
        }
        int i = threadIdx.x;                    // 256 threads, 256 bias elements
        sBias[((i & 1) << 7) + (i >> 1)] = bias[i];
    }
    __syncthreads();

    const int wave  = threadIdx.x >> 5;         // 8 waves per block (wave32)
    const int lane  = threadIdx.x & 31;
    const int lrow  = lane & 15;                // A: M index / B,D: N index
    const int lhalf = lane >> 4;                // K-half selector for A/B frags

    for (int tile = blockIdx.x * 8 + wave; tile < tiles; tile += gridDim.x * 8) {
        const int mBase = tile << 4;

        // ---- Preload all 32 A fragments (16x4 each): lane holds
        // state[mBase+lrow][k0 + 2*lhalf + {0,1}]  -> global_load_b64, NT (read-once)
        const float* aPtr = state + (size_t)(mBase + lrow) * 128 + 2 * lhalf;
        v2f a[32];
#pragma unroll
        for (int s = 0; s < 32; ++s)
            a[s] = __builtin_nontemporal_load((const v2f*)(aPtr + 4 * s));

        const float* bPtr = &sW[lrow * LDS_STRIDE + 2 * lhalf];

        // ---- Two N-tile groups of 8 (group 0 -> delta, group 1 -> var).
        // Sequential groups keep only 8 accumulators (64 VGPRs) live at once.
#pragma unroll 1
        for (int g = 0; g < 2; ++g) {
            const float* bGroup = bPtr + (g * 128) * LDS_STRIDE;
            float* base = g ? outVar : outDelta;

            // Init accumulators with bias (same value for all 16 rows of a tile)
            v8f acc[8];
#pragma unroll
            for (int t = 0; t < 8; ++t) {
                float bv = sBias[g * 128 + t * 16 + lrow];
#pragma unroll
                for (int r = 0; r < 8; ++r) acc[t][r] = bv;
            }

            // K loop: 32 steps of K=4; B-frag lane holds
            // sW[(g*128 + t*16 + lrow)][k0 + 2*lhalf + {0,1}]  -> ds_load_b64
#pragma unroll
            for (int s = 0; s < 32; ++s) {
#pragma unroll
                for (int t = 0; t < 8; ++t) {
                    v2f bf = *(const v2f*)(bGroup + (t * 16) * LDS_STRIDE + 4 * s);
                    acc[t] = __builtin_amdgcn_wmma_f32_16x16x4_f32(
                        false, a[s], false, bf, (short)0, acc[t], false, false);
                }
            }

            // Store: D layout: VGPR r, lanes0-15 -> M=r, lanes16-31 -> M=r+8, N=lrow.
            // 16 lanes write 64B contiguous; NT stores keep L2 for state/W.
#pragma unroll
            for (int t = 0; t < 8; ++t) {
                int col = (t << 4) + lrow;
#pragma unroll
                for (int r = 0; r < 8; ++r) {
                    int row = mBase + r + (lhalf << 3);
                    __builtin_nontemporal_store(acc[t][r], base + (size_t)row * 128 + col);
                }
            }
        }
    }
}

extern "C" void kernel_launch(void* const* d_in, const int* in_sizes, int n_in,
                              void* d_out, int out_size, void* d_ws, size_t ws_size,
                              hipStream_t stream) {
    const float* state = (const float*)d_in[0];   // (B,128) f32
    const float* W     = (const float*)d_in[1];   // (128,2,128) f32
    const float* bias  = (const float*)d_in[2];   // (128,2) f32
    float* out = (float*)d_out;                   // delta (B*128) then var (B*128)

    int Brows = in_sizes[0] / 128;
    int tiles = Brows >> 4;                       // 8192 for B=131072

    int blocks = (tiles + 15) >> 4;               // ~2 M-tiles per wave-slot
    if (blocks > 512) blocks = 512;
    if (blocks < 1)  blocks = 1;

    size_t shmem = (size_t)(256 * LDS_STRIDE + 256) * sizeof(float);  // ~133 KB

    flow_gemm_f32_wmma<<<blocks, 256, shmem, stream>>>(
        state, W, bias, out, out + (size_t)Brows * 128, tiles);
}